// ISDALoss_3023656976548
// MI455X (gfx1250) — compile-verified
//
#include <hip/hip_runtime.h>
#include <hip/hip_bf16.h>
#include <math.h>

#define C_CLASSES 256
#define A_DIM     512
#define N_SAMP    128

typedef __attribute__((ext_vector_type(2))) float v2f;
typedef __attribute__((ext_vector_type(8))) float v8f;

// ---------------------------------------------------------------------------
// K1: per-class mean + counts (gather form). grid = (C, 2): y==0 -> source, 1 -> target.
// Label loads are wave-uniform (scalar path); feature loads coalesce across lanes.
// ---------------------------------------------------------------------------
__global__ void class_stats_kernel(const float* __restrict__ s_feat,
                                   const float* __restrict__ t_feat,
                                   const int* __restrict__ ts,
                                   const int* __restrict__ tt,
                                   float* __restrict__ ave,
                                   float* __restrict__ tave,
                                   float* __restrict__ counts,
                                   float* __restrict__ tcounts) {
    const int c   = blockIdx.x;
    const int sel = blockIdx.y;
    const float* F   = sel ? t_feat  : s_feat;
    const int*   L   = sel ? tt      : ts;
    float*       AVE = sel ? tave    : ave;
    float*       CNT = sel ? tcounts : counts;

    int cnt = 0;
    for (int n = 0; n < N_SAMP; ++n) cnt += (L[n] == c) ? 1 : 0;
    if (threadIdx.x == 0) CNT[c] = (float)cnt;

    const float inv = 1.0f / fmaxf((float)cnt, 1.0f);
    for (int a = threadIdx.x; a < A_DIM; a += blockDim.x) {
        float s = 0.0f;
        for (int n = 0; n < N_SAMP; ++n) {
            if (L[n] == c) s += F[n * A_DIM + a];
        }
        AVE[c * A_DIM + a] = s * inv;
    }
}

// ---------------------------------------------------------------------------
// K2a: U[n,a] = s_features[n,a] - ave[target_s[n], a]
// ---------------------------------------------------------------------------
__global__ void center_kernel(const float* __restrict__ s_feat,
                              const int* __restrict__ ts,
                              const float* __restrict__ ave,
                              float* __restrict__ U) {
    const int i = blockIdx.x * blockDim.x + threadIdx.x;      // < N_SAMP*A_DIM
    const int n = i / A_DIM;
    const int a = i - n * A_DIM;
    U[i] = s_feat[i] - ave[ts[n] * A_DIM + a];
}

// ---------------------------------------------------------------------------
// K2b: G = 0.5*(ave + t_ave)   (empty-class means are exactly 0 -> masks are identity)
// ---------------------------------------------------------------------------
__global__ void gmix_kernel(const float* __restrict__ a,
                            const float* __restrict__ b,
                            float* __restrict__ g) {
    const int i = blockIdx.x * blockDim.x + threadIdx.x;      // < C_CLASSES*A_DIM
    g[i] = 0.5f * (a[i] + b[i]);
}

// ---------------------------------------------------------------------------
// K3: NT fp32 GEMM on the matrix pipe. One wave32 computes one 16x16 tile of
//   Cm[M,N] = Am[M,K] * Bm[N,K]^T (+ bias[col]), via V_WMMA_F32_16X16X4_F32.
// A 16x4 fp32 layout (ISA 7.12.2): lanes 0-15 hold M=lane, VGPR{0,1}=K{0,1};
// lanes 16-31 hold K{2,3}. B mirrors with lane = N. C/D: VGPR v -> row v (lanes
// 0-15) / row v+8 (lanes 16-31), col = lane&15. No divergence; EXEC all ones.
// ---------------------------------------------------------------------------
__global__ void gemm_nt_wmma(const float* __restrict__ Am,
                             const float* __restrict__ Bm,
                             const float* __restrict__ bias,
                             float* __restrict__ Cm,
                             int Kdim, int ldc) {
    const int lane = threadIdx.x;          // 0..31, one wave per block
    const int half = lane >> 4;            // 0: K pair {0,1}, 1: K pair {2,3}
    const int mrow = lane & 15;

    const int tm = blockIdx.x * 16;
    const int tn = blockIdx.y * 16;

    const v2f* pa = (const v2f*)(Am + (size_t)(tm + mrow) * Kdim);
    const v2f* pb = (const v2f*)(Bm + (size_t)(tn + mrow) * Kdim);

    v8f acc = {0.f, 0.f, 0.f, 0.f, 0.f, 0.f, 0.f, 0.f};

    const int k2end = Kdim >> 1;           // K in units of float2
    for (int k2 = 0; k2 < k2end; k2 += 2) {  // 4 K-values per WMMA
        v2f a = pa[k2 + half];
        v2f b = pb[k2 + half];
        acc = __builtin_amdgcn_wmma_f32_16x16x4_f32(
            /*neg_a=*/false, a, /*neg_b=*/false, b,
            /*c_mod=*/(short)0, acc, /*reuse_a=*/false, /*reuse_b=*/false);
    }

    const float bv = bias ? bias[tn + mrow] : 0.0f;
#pragma unroll
    for (int v = 0; v < 8; ++v) {
        const int row = tm + v + 8 * half;
        Cm[(size_t)row * ldc + (tn + mrow)] = acc[v] + bv;
    }
}

// ---------------------------------------------------------------------------
// K4: S[t,c] = (1/max(counts[t],1)) * sum_{m: ts[m]==t} (P[m,c]-P[m,t])^2
// ---------------------------------------------------------------------------
__global__ void sigma_kernel(const float* __restrict__ P,
                             const int* __restrict__ ts,
                             const float* __restrict__ counts,
                             float* __restrict__ S) {
    const int t = blockIdx.x;
    const int c = threadIdx.x;
    float acc = 0.0f;
    for (int m = 0; m < N_SAMP; ++m) {
        if (ts[m] == t) {                   // wave-uniform branch
            const float pt = P[m * C_CLASSES + t];      // broadcast
            const float d  = P[m * C_CLASSES + c] - pt; // coalesced
            acc += d * d;
        }
    }
    S[t * C_CLASSES + c] = acc / fmaxf(counts[t], 1.0f);
}

// ---------------------------------------------------------------------------
// K5: per-sample -log_softmax picked at target:  rloss[n] = lse(z) - z[tc]
//     z[c] = Y[tc,c] + 0.25 * S[tc,c]     (0.5 * RATIO_ALPHA = 0.25)
// ---------------------------------------------------------------------------
__global__ void row_loss_kernel(const float* __restrict__ Y,
                                const float* __restrict__ S,
                                const int* __restrict__ tt,
                                float* __restrict__ rloss) {
    __shared__ float sm[C_CLASSES];
    __shared__ float ztgt;
    const int n  = blockIdx.x;
    const int c  = threadIdx.x;
    const int tc = tt[n];

    const float z = Y[tc * C_CLASSES + c] + 0.25f * S[tc * C_CLASSES + c];
    if (c == tc) ztgt = z;
    sm[c] = z;
    __syncthreads();

    for (int s = C_CLASSES / 2; s > 0; s >>= 1) {
        if (c < s) sm[c] = fmaxf(sm[c], sm[c + s]);
        __syncthreads();
    }
    const float mx = sm[0];
    __syncthreads();

    sm[c] = __expf(z - mx);
    __syncthreads();
    for (int s = C_CLASSES / 2; s > 0; s >>= 1) {
        if (c < s) sm[c] += sm[c + s];
        __syncthreads();
    }
    if (c == 0) rloss[n] = mx + __logf(sm[0]) - ztgt;
}

// ---------------------------------------------------------------------------
// K6: final mean over N samples -> scalar loss (no atomics; deterministic)
// ---------------------------------------------------------------------------
__global__ void mean_kernel(const float* __restrict__ rloss,
                            float* __restrict__ out) {
    __shared__ float sm[N_SAMP];
    const int i = threadIdx.x;
    sm[i] = rloss[i];
    __syncthreads();
    for (int s = N_SAMP / 2; s > 0; s >>= 1) {
        if (i < s) sm[i] += sm[i + s];
        __syncthreads();
    }
    if (i == 0) out[0] = sm[0] * (1.0f / (float)N_SAMP);
}

// ---------------------------------------------------------------------------
extern "C" void kernel_launch(void* const* d_in, const int* in_sizes, int n_in,
                              void* d_out, int out_size, void* d_ws, size_t ws_size,
                              hipStream_t stream) {
    const float* fc_w = (const float*)d_in[0];   // (2C, A)
    const float* fc_b = (const float*)d_in[1];   // (2C,)
    const float* s_f  = (const float*)d_in[2];   // (N, A)
    const float* t_f  = (const float*)d_in[3];   // (N, A)
    const int*   ts   = (const int*)d_in[4];     // (N,)
    const int*   tt   = (const int*)d_in[5];     // (N,)
    float* out = (float*)d_out;

    float* ws = (float*)d_ws;
    float* ave    = ws;  ws += C_CLASSES * A_DIM;     // 131072
    float* tave   = ws;  ws += C_CLASSES * A_DIM;     // 131072
    float* counts = ws;  ws += C_CLASSES;             // 256
    float* tcnt   = ws;  ws += C_CLASSES;             // 256
    float* U      = ws;  ws += N_SAMP * A_DIM;        // 65536
    float* G      = ws;  ws += C_CLASSES * A_DIM;     // 131072
    float* P      = ws;  ws += N_SAMP * C_CLASSES;    // 32768
    float* Y      = ws;  ws += C_CLASSES * C_CLASSES; // 65536
    float* S      = ws;  ws += C_CLASSES * C_CLASSES; // 65536
    float* rl     = ws;                               // 128  (total ~2.4 MB)

    class_stats_kernel<<<dim3(C_CLASSES, 2), 256, 0, stream>>>(
        s_f, t_f, ts, tt, ave, tave, counts, tcnt);

    center_kernel<<<(N_SAMP * A_DIM) / 256, 256, 0, stream>>>(s_f, ts, ave, U);
    gmix_kernel<<<(C_CLASSES * A_DIM) / 256, 256, 0, stream>>>(ave, tave, G);

    // P = U @ W[:C]^T          (M=128, N=256, K=512)
    gemm_nt_wmma<<<dim3(N_SAMP / 16, C_CLASSES / 16), 32, 0, stream>>>(
        U, fc_w, nullptr, P, A_DIM, C_CLASSES);
    // Y = G @ W[:C]^T + b[:C]  (M=256, N=256, K=512)
    gemm_nt_wmma<<<dim3(C_CLASSES / 16, C_CLASSES / 16), 32, 0, stream>>>(
        G, fc_w, fc_b, Y, A_DIM, C_CLASSES);

    sigma_kernel<<<C_CLASSES, C_CLASSES, 0, stream>>>(P, ts, counts, S);
    row_loss_kernel<<<N_SAMP, C_CLASSES, 0, stream>>>(Y, S, tt, rl);
    mean_kernel<<<1, N_SAMP, 0, stream>>>(rl, out);
}